// QuantizedAdaptiveAvgPool2d_45861660787301
// MI455X (gfx1250) — compile-verified
//
#include <hip/hip_runtime.h>

typedef __attribute__((ext_vector_type(2))) float v2f;
typedef __attribute__((ext_vector_type(4))) float v4f;
typedef __attribute__((ext_vector_type(8))) float v8f;

#define IN_H 224
#define IN_W 224
#define OUT_D 7
#define WIN 32          // 224/7
#define NTHREADS 224    // 7 waves of 32

// One block per (b,c) plane. Phase 1: bandwidth-bound streaming reduction
// (float4 NT loads, fp32 register accumulators). Phase 2: exact-fp32 WMMA
// epilogue: D = A x ones(4x16) + C sums 4 K-values/row/issue, chained 8x.
__global__ __launch_bounds__(NTHREADS)
void adaptive_avgpool_7x7_kernel(const float* __restrict__ x,
                                 float* __restrict__ out) {
    // part[row][k]: row = band*7 + ow (0..48), k = c8*4 + r_off (0..31).
    // Stride 33 floats -> conflict-free column reads. Rows 49..63 are the
    // zero pad consumed by the last (partially valid) WMMA chunk.
    __shared__ float part[64][33];

    const int t     = threadIdx.x;
    const int r_off = t / 56;   // 0..3 : row phase within a band
    const int col4  = t % 56;   // 0..55: float4 column group
    const float* __restrict__ plane =
        x + (size_t)blockIdx.x * (IN_H * IN_W);

    // Zero the whole LDS tile (covers the pad rows read by chunk 3).
    for (int idx = t; idx < 64 * 33; idx += NTHREADS)
        (&part[0][0])[idx] = 0.0f;

    // ---- Phase 1: per-thread fp32 accumulation over 8 rows x 4 cols per band
    const float* __restrict__ colp = plane + col4 * 4;
    float psum[OUT_D];
    #pragma unroll
    for (int b = 0; b < OUT_D; ++b) {
        v4f s = {0.0f, 0.0f, 0.0f, 0.0f};
        #pragma unroll
        for (int i = 0; i < 8; ++i) {
            const int row = b * WIN + i * 4 + r_off;
            const v4f v = __builtin_nontemporal_load(
                (const v4f*)(colp + (size_t)row * IN_W));
            s += v;
        }
        psum[b] = s.x + s.y + s.z + s.w;
    }

    __syncthreads();  // zero-fill complete before partial scatter

    #pragma unroll
    for (int b = 0; b < OUT_D; ++b) {
        const int row = b * OUT_D + (col4 >> 3);       // output index 0..48
        const int k   = ((col4 & 7) << 2) | r_off;     // partial index 0..31
        part[row][k] = psum[b];
    }

    __syncthreads();

    // ---- Phase 2: WMMA f32 16x16x4 segmented reduction, waves 0..3
    const int wave = t >> 5;
    const int lane = t & 31;
    if (wave < 4) {
        const int base = wave * 16;            // first output row of this chunk
        const int m    = lane & 15;            // M = lane % 16
        const int koff = (lane >> 4) << 1;     // lanes 0-15: K+0/1, 16-31: K+2/3

        v8f c = {0.f, 0.f, 0.f, 0.f, 0.f, 0.f, 0.f, 0.f};
        v2f ones; ones.x = 1.0f; ones.y = 1.0f;   // B = ones(4x16)

        #pragma unroll
        for (int kb = 0; kb < 8; ++kb) {
            v2f a;
            a.x = part[base + m][kb * 4 + koff];
            a.y = part[base + m][kb * 4 + koff + 1];
            // D[m][n] = sum_k A[m][k] + C[m][n]  (exact fp32 row-sum)
            c = __builtin_amdgcn_wmma_f32_16x16x4_f32(
                    /*neg_a=*/false, a, /*neg_b=*/false, ones,
                    /*c_mod=*/(short)0, c, /*reuse_a=*/false, /*reuse_b=*/false);
        }

        // D layout: VGPR i = rows base+i (lanes 0-15) / base+8+i (lanes 16-31);
        // all N columns identical (ones-B), so lane 0 / lane 16 suffice.
        const float scale = 1.0f / (float)(WIN * WIN);
        float* __restrict__ o = out + (size_t)blockIdx.x * (OUT_D * OUT_D);
        if (lane == 0) {
            #pragma unroll
            for (int i = 0; i < 8; ++i) {
                const int row = base + i;
                if (row < OUT_D * OUT_D) o[row] = c[i] * scale;
            }
        } else if (lane == 16) {
            #pragma unroll
            for (int i = 0; i < 8; ++i) {
                const int row = base + 8 + i;
                if (row < OUT_D * OUT_D) o[row] = c[i] * scale;
            }
        }
    }
}

extern "C" void kernel_launch(void* const* d_in, const int* in_sizes, int n_in,
                              void* d_out, int out_size, void* d_ws, size_t ws_size,
                              hipStream_t stream) {
    (void)n_in; (void)d_ws; (void)ws_size; (void)out_size;
    const float* x = (const float*)d_in[0];
    float* out = (float*)d_out;
    const int planes = in_sizes[0] / (IN_H * IN_W);   // 16*256 = 4096
    adaptive_avgpool_7x7_kernel<<<planes, NTHREADS, 0, stream>>>(x, out);
}